// GraphSAGELayer_55748675502376
// MI455X (gfx1250) — compile-verified
//
#include <hip/hip_runtime.h>

#define BB   4
#define NN   2048
#define CC   256
#define OUTC 256
#define SS   25
#define K2   (2 * CC)      // 512
#define ROWS (BB * NN)     // 8192
#define LDS_STRIDE 516     // 512 + 4 pad -> row m occupies bank (4m + k) % 64, conflict-free

typedef float v2f __attribute__((ext_vector_type(2)));
typedef float v8f __attribute__((ext_vector_type(8)));

// ---------------------------------------------------------------------------
// Kernel 1: per-node ordered neighbor selection (lowest index first, up to S)
// + full degree count. One wave32 per row of A. Bound by the 67MB HBM scan.
// ---------------------------------------------------------------------------
__global__ __launch_bounds__(256) void sage_select(const int* __restrict__ A,
                                                   int* __restrict__ nbr,
                                                   int* __restrict__ deg)
{
    const int wave = threadIdx.x >> 5;
    const int lane = threadIdx.x & 31;
    const int row  = blockIdx.x * 8 + wave;
    if (row >= ROWS) return;

    const int4* arow = (const int4*)(A + (long long)row * NN);
    int total = 0;

    for (int it = 0; it < NN / 128; ++it) {
        int4 v = arow[it * 32 + lane];
        int m0 = (v.x != 0), m1 = (v.y != 0), m2 = (v.z != 0), m3 = (v.w != 0);
        int cnt = m0 + m1 + m2 + m3;

        // inclusive prefix over 32 lanes
        int incl = cnt;
        #pragma unroll
        for (int off = 1; off < 32; off <<= 1) {
            int t = __shfl_up(incl, off);
            if (lane >= off) incl += t;
        }
        int base = total + (incl - cnt);   // ordered global position of this lane's 1st hit
        int colbase = it * 128 + lane * 4;

        int r = 0;
        if (m0) { if (base + r < SS) nbr[row * SS + base + r] = colbase + 0; r++; }
        if (m1) { if (base + r < SS) nbr[row * SS + base + r] = colbase + 1; r++; }
        if (m2) { if (base + r < SS) nbr[row * SS + base + r] = colbase + 2; r++; }
        if (m3) { if (base + r < SS) nbr[row * SS + base + r] = colbase + 3; r++; }

        total += __shfl(incl, 31);         // wave-uniform running degree
    }
    if (lane == 0) deg[row] = total;
}

// ---------------------------------------------------------------------------
// Kernel 2: fused max-aggregate + concat + GEMM(2C x OUT) + bias + ReLU.
// One block = 16 node rows. h tile (16 x 512 f32) staged in LDS, then
// V_WMMA_F32_16X16X4_F32 over K=512 with two independent accumulation chains
// per wave (tiles tn=wave and tn=wave+8 share each A fragment).
// ---------------------------------------------------------------------------
__global__ __launch_bounds__(256) void sage_gemm(const float* __restrict__ X,
                                                 const float* __restrict__ W,
                                                 const float* __restrict__ bias,
                                                 const int* __restrict__ nbr,
                                                 const int* __restrict__ deg,
                                                 float* __restrict__ out)
{
    __shared__ float h[16 * LDS_STRIDE];

    const int wave = threadIdx.x >> 5;
    const int lane = threadIdx.x & 31;
    const int row0 = blockIdx.x * 16;

    // ---- Phase A: build h = [X_self | max_agg] for 16 rows (2 rows/wave) ----
    #pragma unroll
    for (int rr = 0; rr < 2; ++rr) {
        const int li    = wave * 2 + rr;        // local row 0..15
        const int row   = row0 + li;            // global row
        const int batch = row >> 11;            // row / 2048
        const float* xb = X + (size_t)batch * NN * CC;
        const float* xs = X + (size_t)row * CC; // self features
        const int ch    = lane * 8;             // 32 lanes * 8 ch = 256

        float4 s0 = *(const float4*)(xs + ch);
        float4 s1 = *(const float4*)(xs + ch + 4);
        *(float4*)&h[li * LDS_STRIDE + ch]     = s0;
        *(float4*)&h[li * LDS_STRIDE + ch + 4] = s1;

        const float NEGINF = -__builtin_inff();
        float4 a0 = make_float4(NEGINF, NEGINF, NEGINF, NEGINF);
        float4 a1 = a0;

        const int d  = deg[row];
        const int nn = d < SS ? d : SS;
        const int* nb = nbr + row * SS;
        for (int t = 0; t < nn; ++t) {
            const float* xn = xb + (size_t)nb[t] * CC + ch;
            float4 v0 = *(const float4*)(xn);
            float4 v1 = *(const float4*)(xn + 4);
            a0.x = fmaxf(a0.x, v0.x); a0.y = fmaxf(a0.y, v0.y);
            a0.z = fmaxf(a0.z, v0.z); a0.w = fmaxf(a0.w, v0.w);
            a1.x = fmaxf(a1.x, v1.x); a1.y = fmaxf(a1.y, v1.y);
            a1.z = fmaxf(a1.z, v1.z); a1.w = fmaxf(a1.w, v1.w);
        }
        if (d <= SS) {   // zero pad participates in the max (also handles d==0)
            a0.x = fmaxf(a0.x, 0.f); a0.y = fmaxf(a0.y, 0.f);
            a0.z = fmaxf(a0.z, 0.f); a0.w = fmaxf(a0.w, 0.f);
            a1.x = fmaxf(a1.x, 0.f); a1.y = fmaxf(a1.y, 0.f);
            a1.z = fmaxf(a1.z, 0.f); a1.w = fmaxf(a1.w, 0.f);
        }
        *(float4*)&h[li * LDS_STRIDE + CC + ch]     = a0;
        *(float4*)&h[li * LDS_STRIDE + CC + ch + 4] = a1;
    }

    __syncthreads();

    // ---- Phase B: WMMA GEMM, fused dual-tile K loop ----
    const int m     = lane & 15;            // A row, B/C column within tile
    const int khalf = (lane >> 4) * 2;      // K sub-pair selector per ISA layout
    const float* hrow = &h[m * LDS_STRIDE];

    const int n0 = wave * 16 + m;           // column for tile tn = wave
    const int n1 = n0 + 128;                // column for tile tn = wave + 8

    v8f acc0 = {};
    v8f acc1 = {};
    for (int k = 0; k < K2; k += 4) {
        // A 16x4 f32: lanes 0-15 -> K={k,k+1}, lanes 16-31 -> K={k+2,k+3}
        v2f a = *(const v2f*)(hrow + k + khalf);
        // B 4x16 f32: VGPR v, lanes 0-15 -> K=k+v, lanes 16-31 -> K=k+v+2
        const float* wr = W + (size_t)(k + khalf) * OUTC;
        v2f b0, b1;
        b0.x = wr[n0];        b0.y = wr[OUTC + n0];
        b1.x = wr[n1];        b1.y = wr[OUTC + n1];
        acc0 = __builtin_amdgcn_wmma_f32_16x16x4_f32(
            false, a, false, b0, (short)0, acc0, false, false);
        acc1 = __builtin_amdgcn_wmma_f32_16x16x4_f32(
            false, a, false, b1, (short)0, acc1, false, false);
    }

    const float bv0 = bias[n0];
    const float bv1 = bias[n1];
    const int mbase = (lane >> 4) * 8;      // C/D: VGPR i -> M = i + 8*(lane/16)
    #pragma unroll
    for (int i = 0; i < 8; ++i) {
        float* orow = out + (size_t)(row0 + mbase + i) * OUTC;
        float v0 = acc0[i] + bv0;
        float v1 = acc1[i] + bv1;
        orow[n0] = v0 > 0.f ? v0 : 0.f;
        orow[n1] = v1 > 0.f ? v1 : 0.f;
    }
}

// ---------------------------------------------------------------------------
extern "C" void kernel_launch(void* const* d_in, const int* in_sizes, int n_in,
                              void* d_out, int out_size, void* d_ws, size_t ws_size,
                              hipStream_t stream)
{
    const int*   A    = (const int*)d_in[0];    // [B,N,N]
    const float* X    = (const float*)d_in[1];  // [B,N,C]
    const float* W    = (const float*)d_in[2];  // [2C,OUT]
    const float* bias = (const float*)d_in[3];  // [OUT]
    float*       out  = (float*)d_out;          // [B,N,OUT]

    int* nbr = (int*)d_ws;                      // ROWS * S indices
    int* deg = nbr + (size_t)ROWS * SS;         // ROWS degrees

    sage_select<<<ROWS / 8, 256, 0, stream>>>(A, nbr, deg);
    sage_gemm<<<ROWS / 16, 256, 0, stream>>>(X, W, bias, nbr, deg, out);
}